// MultiHeadSelfAttention_6038724018819
// MI455X (gfx1250) — compile-verified
//
#include <hip/hip_runtime.h>
#include <stdint.h>

// MI455X / gfx1250, wave32. Memory-bound fused MHSA-capsule kernel:
//   reads  : U_hat 25.7MB (streamed) + W tables 64KB (TDM-staged to LDS once/block)
//   writes : C (205MB) + U_hat_I (205MB) via nontemporal b128 stores
// One wave per row (lane = f). Grid chosen so nw % 32 == 0 -> each wave's
// (d,e) slice of the weight tables is LOOP-INVARIANT and fully hoisted.
// 32-wide reductions: DPP8/DPP16 adds (VALU pipe) + one ds shuffle for cross-16.

typedef float    f32x4 __attribute__((ext_vector_type(4)));
typedef uint32_t u32x4 __attribute__((ext_vector_type(4)));
typedef int      i32x4 __attribute__((ext_vector_type(4)));
typedef int      i32x8 __attribute__((ext_vector_type(8)));

#define ROWS    200704u     // B*S1*S2*NC*DC = 32*14*14*2*16
#define HALF    51380224u   // ROWS * F * I  (elements per output tensor)
#define W_ELEMS 8192u       // NC*DC*F*I
#define NBLK    1024u
#define NWAVES  (NBLK * 8u) // 8192; multiple of 32 (keeps de loop-invariant)

#define DPP8SEL(a,b,c,d,e,f,g,h) \
    ((a)|((b)<<3)|((c)<<6)|((d)<<9)|((e)<<12)|((f)<<15)|((g)<<18)|((h)<<21))

// Pairing permutations: each step pairs every lane with a lane holding the
// complementary partial sum -> valid sum-reduction even though ROW_MIRROR
// is not literally xor-8.
__device__ __forceinline__ float wave_sum32(float v) {
    int o;
    o = __builtin_amdgcn_mov_dpp8(__float_as_int(v), DPP8SEL(1,0,3,2,5,4,7,6)); // xor 1
    v += __int_as_float(o);
    o = __builtin_amdgcn_mov_dpp8(__float_as_int(v), DPP8SEL(2,3,0,1,6,7,4,5)); // xor 2
    v += __int_as_float(o);
    o = __builtin_amdgcn_mov_dpp8(__float_as_int(v), DPP8SEL(4,5,6,7,0,1,2,3)); // xor 4
    v += __int_as_float(o);
    o = __builtin_amdgcn_update_dpp(0, __float_as_int(v), 0x140, 0xf, 0xf, true); // ROW_MIRROR: cross-8
    v += __int_as_float(o);
    v += __shfl_xor(v, 16, 32);                                                  // cross-16
    return v;
}

// TDM: 1D copy of `ndw` dwords from global to LDS (D# per CDNA5 ISA ch.8).
__device__ __forceinline__ void tdm_load_dwords(const float* gsrc,
                                                const float* lds_dst,
                                                uint32_t ndw) {
    uint64_t ga = (uint64_t)(uintptr_t)gsrc;
    uint32_t la = (uint32_t)(uintptr_t)lds_dst;  // flat-LDS addr[31:0] == LDS byte offset
    u32x4 g0 = { 1u,                                   // count=1, user D#, gather off
                 la,                                   // lds_addr
                 (uint32_t)ga,                         // global_addr[31:0]
                 ((uint32_t)(ga >> 32) & 0x01FFFFFFu) | 0x80000000u }; // ga[56:32] | type=2
    i32x8 g1;
    g1[0] = 0x00020000;                       // workgroup_mask=0, data_size=2 (4B), no flags
    g1[1] = (int)((ndw & 0xFFFFu) << 16);     // atomic_addr=0 | tensor_dim0[15:0]
    g1[2] = (int)(((ndw >> 16) & 0xFFFFu) | (1u << 16)); // tensor_dim0[31:16] | tensor_dim1=1
    g1[3] = (int)((ndw & 0xFFFFu) << 16);     // tensor_dim1 hi=0 | tile_dim0=ndw
    g1[4] = 1;                                // tile_dim1=1, tile_dim2=0 (unused)
    g1[5] = (int)ndw;                         // tensor_dim0_stride[31:0]
    g1[6] = (int)((ndw & 0xFFFFu) << 16);     // t0s[47:32]=0 | tensor_dim1_stride[15:0]
    g1[7] = (int)(ndw >> 16);                 // tensor_dim1_stride[47:16]
    i32x4 gz = { 0, 0, 0, 0 };
#if defined(__clang_major__) && (__clang_major__ >= 23)
    i32x8 gz8 = { 0, 0, 0, 0, 0, 0, 0, 0 };
    __builtin_amdgcn_tensor_load_to_lds(g0, g1, gz, gz, gz8, 0);
#else
    __builtin_amdgcn_tensor_load_to_lds(g0, g1, gz, gz, 0);
#endif
}

__global__ __launch_bounds__(256)
void mhsa_fused_kernel(const float* __restrict__ U_hat,
                       const float* __restrict__ W_t,
                       const float* __restrict__ W_affine,
                       float* __restrict__ out) {
    __shared__ __align__(16) float lds[2 * W_ELEMS + 256];
    float* lds_wa = lds;                 // W_affine[de,f,i]  (8192 f32)
    float* lds_wt = lds + W_ELEMS;       // W_t[de,f,i]       (8192 f32)
    float* lds_mw = lds + 2 * W_ELEMS;   // max_f W_t[de,:,i] (256 f32)

    // Wave 0 DMAs both weight tables into LDS via the Tensor Data Mover.
    if (threadIdx.x < 32) {
        tdm_load_dwords(W_affine, lds_wa, W_ELEMS);
        tdm_load_dwords(W_t,      lds_wt, W_ELEMS);
        __builtin_amdgcn_s_wait_tensorcnt(0);
    }
    __syncthreads();

    // Precompute mw[de,i] = max_f W_t[de,f,i] once per block (for stable softmax).
    {
        const int t = (int)threadIdx.x;            // t = de*8 + i
        const float* p = &lds_wt[(t >> 3) * 256 + (t & 7)];
        float m = p[0];
#pragma unroll
        for (int f = 1; f < 32; ++f) m = fmaxf(m, p[f * 8]);
        lds_mw[t] = m;
    }
    __syncthreads();

    const unsigned lane = threadIdx.x & 31u;                  // lane == f
    const unsigned wid0 = blockIdx.x * 8u + (threadIdx.x >> 5);
    // Row stride NWAVES is a multiple of 32 -> de = r & 31 is constant per wave:
    const unsigned de   = wid0 & 31u;                         // d*16 + e
    const unsigned wb   = de * 256u + lane * 8u;

    // Hoist ALL weight-table reads out of the row loop.
    const f32x4 wa0 = *(const f32x4*)&lds_wa[wb];
    const f32x4 wa1 = *(const f32x4*)&lds_wa[wb + 4];
    const f32x4 wt0 = *(const f32x4*)&lds_wt[wb];
    const f32x4 wt1 = *(const f32x4*)&lds_wt[wb + 4];
    const f32x4 mw0 = *(const f32x4*)&lds_mw[de * 8u];
    const f32x4 mw1 = *(const f32x4*)&lds_mw[de * 8u + 4];
    // exp(d*wt - d*mw) == exp(d*(wt-mw)); td is loop-invariant and <= 0.
    const f32x4 td0 = wt0 - mw0;
    const f32x4 td1 = wt1 - mw1;
    const float td[8] = { td0.x, td0.y, td0.z, td0.w, td1.x, td1.y, td1.z, td1.w };

    for (unsigned r = wid0; r < ROWS; r += NWAVES) {
        __builtin_prefetch(&U_hat[(size_t)(r + NWAVES) * 32u + lane], 0, 3);

        const float u = U_hat[(size_t)r * 32u + lane];
        const f32x4 uh0 = u * wa0;                 // U_hat_I for i=0..3
        const f32x4 uh1 = u * wa1;                 // U_hat_I for i=4..7
        const float uh[8] = { uh0.x, uh0.y, uh0.z, uh0.w, uh1.x, uh1.y, uh1.z, uh1.w };
        float c[8];
#pragma unroll
        for (int i = 0; i < 8; ++i) {
            // diag = (sum_f uh^2) / sqrt(16); diag >= 0, so the softmax max is
            // diag*max_f(wt) exactly (fp rounding is monotone under scaling).
            const float d  = wave_sum32(uh[i] * uh[i]) * 0.25f;
            const float ex = __expf(d * td[i]);    // stable softmax numerator
            const float s  = wave_sum32(ex);
            c[i] = __fdividef(ex, s);
        }

        // lane writes 8 consecutive floats -> two fully-coalesced b128 NT stores per tensor
        const size_t ob = (size_t)r * 256u + lane * 8u;
        const f32x4 c0 = { c[0], c[1], c[2], c[3] };
        const f32x4 c1 = { c[4], c[5], c[6], c[7] };
        __builtin_nontemporal_store(c0,  (f32x4*)&out[ob]);            // C
        __builtin_nontemporal_store(c1,  (f32x4*)&out[ob + 4]);
        __builtin_nontemporal_store(uh0, (f32x4*)&out[HALF + ob]);     // U_hat_I
        __builtin_nontemporal_store(uh1, (f32x4*)&out[HALF + ob + 4]);
    }
}

extern "C" void kernel_launch(void* const* d_in, const int* in_sizes, int n_in,
                              void* d_out, int out_size, void* d_ws, size_t ws_size,
                              hipStream_t stream) {
    (void)in_sizes; (void)n_in; (void)out_size; (void)d_ws; (void)ws_size;
    const float* U_hat    = (const float*)d_in[0];
    const float* W_t      = (const float*)d_in[1];
    const float* W_affine = (const float*)d_in[2];
    float* out = (float*)d_out;
    // 1024 blocks x 8 waves = 8192 waves; ~24.5 rows each (grid-stride).
    // TDM weight-staging traffic: 1024 * 64KB = 64MB of L2 reads (tables L2-resident).
    hipLaunchKernelGGL(mhsa_fused_kernel, dim3(NBLK), dim3(256), 0, stream,
                       U_hat, W_t, W_affine, out);
}